// TeacherEnsemble_50903952392532
// MI455X (gfx1250) — compile-verified
//
#include <hip/hip_runtime.h>
#include <hip/hip_bf16.h>
#include <math.h>

#define BATCH        2048
#define OBS_DIM      48
#define H0D          512
#define H1D          256
#define H2D          128
#define ACT_DIM      12
#define NUM_TEACHERS 64

typedef __attribute__((ext_vector_type(2))) float v2f;
typedef __attribute__((ext_vector_type(8))) float v8f;

// ---------------------------------------------------------------------------
// Pass 1: bin samples by teacher (lists live in d_ws).
// Slot order inside a teacher is atomic-nondeterministic, but each sample's
// output depends only on its own column -> final d_out is deterministic.
// ---------------------------------------------------------------------------
__global__ void te_zero_counts(int* __restrict__ counts) {
    if (threadIdx.x < NUM_TEACHERS) counts[threadIdx.x] = 0;
}

__global__ void te_bin(const int* __restrict__ tidx, int* __restrict__ counts,
                       int* __restrict__ lists) {
    int i = blockIdx.x * blockDim.x + threadIdx.x;
    if (i < BATCH) {
        int t    = tidx[i];
        int slot = atomicAdd(counts + t, 1);
        lists[t * BATCH + slot] = i;
    }
}

__device__ __forceinline__ float elu_fast(float r) {
    float e = __expf(r) - 1.0f;          // single v_exp_f32, branchless select
    return r > 0.0f ? r : e;
}

// ---------------------------------------------------------------------------
// One fused-layer pass: out[MO][16] = ELU(W(MO x MI) * in[MI][16] + b)
// Two M-tiles per wave iteration: shared B-fragment, two independent
// WMMA accumulation chains (ILP), bias prefetched as 2x float4 per tile.
//
// A-frag (16x4 f32): lane holds W[m0+m][kk+kb], W[m0+m][kk+kb+1]
//                    with m = lane&15, kb = 2*(lane>=16)
// B-frag (4x16 f32): lane holds in[kk+kb][n], in[kk+kb+1][n], n = lane&15
// C/D   (16x16 f32): VGPR v <-> row m0 + v + 8*(lane>=16), col = lane&15
// ---------------------------------------------------------------------------
template <int MO, int MI>
__device__ __forceinline__ void layer_fw(const float* __restrict__ W,
                                         const float* __restrict__ B,
                                         const float* __restrict__ in_lds,
                                         float* __restrict__ out_lds,
                                         int wave, int lane) {
    const int m  = lane & 15;
    const int hi = lane >> 4;
    const int kb = hi * 2;
    constexpr int NT = MO / 16;
    static_assert(NT % 8 == 0, "need pairs of M-tiles per wave");
    for (int mt = wave; mt < NT; mt += 8) {          // pair {mt, mt+4}
        const int m0 = mt * 16;
        const int m1 = (mt + 4) * 16;
        const float* __restrict__ wr0 = W + (size_t)(m0 + m) * MI;
        const float* __restrict__ wr1 = W + (size_t)(m1 + m) * MI;
        // Hoisted bias loads (16B-aligned, contiguous 8 floats per lane-half);
        // latency hides under the K loop.
        const float4 p00 = ((const float4*)(B + m0 + hi * 8))[0];
        const float4 p01 = ((const float4*)(B + m0 + hi * 8))[1];
        const float4 p10 = ((const float4*)(B + m1 + hi * 8))[0];
        const float4 p11 = ((const float4*)(B + m1 + hi * 8))[1];
        v8f acc0 = {};
        v8f acc1 = {};
#pragma unroll 4
        for (int kk = 0; kk < MI; kk += 4) {
            v2f a0, a1, b;
            a0.x = wr0[kk + kb];
            a0.y = wr0[kk + kb + 1];
            a1.x = wr1[kk + kb];
            a1.y = wr1[kk + kb + 1];
            b.x  = in_lds[(kk + kb) * 16 + m];
            b.y  = in_lds[(kk + kb + 1) * 16 + m];
            acc0 = __builtin_amdgcn_wmma_f32_16x16x4_f32(
                false, a0, false, b, (short)0, acc0, false, false);
            acc1 = __builtin_amdgcn_wmma_f32_16x16x4_f32(
                false, a1, false, b, (short)0, acc1, false, false);
        }
        const float bias0[8] = {p00.x, p00.y, p00.z, p00.w, p01.x, p01.y, p01.z, p01.w};
        const float bias1[8] = {p10.x, p10.y, p10.z, p10.w, p11.x, p11.y, p11.z, p11.w};
#pragma unroll
        for (int v = 0; v < 8; ++v) {
            out_lds[(m0 + v + hi * 8) * 16 + m] = elu_fast(acc0[v] + bias0[v]);
            out_lds[(m1 + v + hi * 8) * 16 + m] = elu_fast(acc1[v] + bias1[v]);
        }
    }
}

// ---------------------------------------------------------------------------
// Main kernel: block = (teacher, 16-sample tile), 128 threads = 4 wave32.
// Activations staged in LDS between layers (~59 KB / block, 320 KB per WGP).
// ---------------------------------------------------------------------------
__global__ void __launch_bounds__(128)
teacher_mlp(const float* __restrict__ obs,
            const int* __restrict__ counts, const int* __restrict__ lists,
            const float* __restrict__ w0, const float* __restrict__ b0,
            const float* __restrict__ w2, const float* __restrict__ b2,
            const float* __restrict__ w4, const float* __restrict__ b4,
            const float* __restrict__ w6, const float* __restrict__ b6,
            float* __restrict__ out) {
    __shared__ float xin[OBS_DIM * 16];
    __shared__ float h0[H0D * 16];
    __shared__ float h1[H1D * 16];
    __shared__ float h2[H2D * 16];
    __shared__ int   sidx[16];

    const int t    = blockIdx.x;
    const int tile = blockIdx.y;
    const int cnt  = counts[t];
    if (tile * 16 >= cnt) return;                     // uniform exit (pre-barrier)

    const int lane = threadIdx.x & 31;
    const int wave = threadIdx.x >> 5;

    if (threadIdx.x < 16) {
        int p = tile * 16 + threadIdx.x;
        sidx[threadIdx.x] = (p < cnt) ? lists[t * BATCH + p] : -1;
    }
    __syncthreads();

    // Stage obs tile into LDS, k-major [K][16]; pad missing columns with 0.
    for (int e = threadIdx.x; e < OBS_DIM * 16; e += 128) {
        int k = e >> 4, n = e & 15;
        int s = sidx[n];
        xin[k * 16 + n] = (s >= 0) ? obs[(size_t)s * OBS_DIM + k] : 0.0f;
    }
    __syncthreads();

    layer_fw<H0D, OBS_DIM>(w0 + (size_t)t * H0D * OBS_DIM, b0 + t * H0D, xin, h0, wave, lane);
    __syncthreads();
    layer_fw<H1D, H0D>(w2 + (size_t)t * H1D * H0D, b2 + t * H1D, h0, h1, wave, lane);
    __syncthreads();
    layer_fw<H2D, H1D>(w4 + (size_t)t * H2D * H1D, b4 + t * H2D, h1, h2, wave, lane);
    __syncthreads();

    // Final layer: 12x128, single padded M-tile, wave 0 only. tanh + store.
    if (wave == 0) {
        const float* __restrict__ W = w6 + (size_t)t * ACT_DIM * H2D;
        const float* __restrict__ B = b6 + t * ACT_DIM;
        const int m  = lane & 15;
        const int hi = lane >> 4;
        const int kb = hi * 2;
        v8f acc = {};
#pragma unroll 4
        for (int kk = 0; kk < H2D; kk += 4) {
            v2f a, b;
            if (m < ACT_DIM) {                        // pad rows 12..15 with 0
                a.x = W[m * H2D + kk + kb];
                a.y = W[m * H2D + kk + kb + 1];
            } else {
                a.x = 0.0f;
                a.y = 0.0f;
            }
            b.x = h2[(kk + kb) * 16 + m];
            b.y = h2[(kk + kb + 1) * 16 + m];
            acc = __builtin_amdgcn_wmma_f32_16x16x4_f32(
                false, a, false, b, (short)0, acc, false, false);
        }
        const int col = lane & 15;
        const int s   = sidx[col];
        if (s >= 0) {
#pragma unroll
            for (int v = 0; v < 8; ++v) {
                int row = v + hi * 8;
                if (row < ACT_DIM)
                    out[(size_t)s * ACT_DIM + row] = tanhf(acc[v] + B[row]);
            }
        }
    }
}

// ---------------------------------------------------------------------------
extern "C" void kernel_launch(void* const* d_in, const int* in_sizes, int n_in,
                              void* d_out, int out_size, void* d_ws, size_t ws_size,
                              hipStream_t stream) {
    (void)in_sizes; (void)n_in; (void)out_size; (void)ws_size;
    const float* obs  = (const float*)d_in[0];
    const int*   tidx = (const int*)d_in[1];
    const float* w0   = (const float*)d_in[2];
    const float* b0   = (const float*)d_in[3];
    const float* w2   = (const float*)d_in[4];
    const float* b2   = (const float*)d_in[5];
    const float* w4   = (const float*)d_in[6];
    const float* b4   = (const float*)d_in[7];
    const float* w6   = (const float*)d_in[8];
    const float* b6   = (const float*)d_in[9];
    float*       out  = (float*)d_out;

    int* counts = (int*)d_ws;            // 64 ints
    int* lists  = counts + NUM_TEACHERS; // 64 * 2048 ints

    te_zero_counts<<<1, 64, 0, stream>>>(counts);
    te_bin<<<BATCH / 256, 256, 0, stream>>>(tidx, counts, lists);

    dim3 grid(NUM_TEACHERS, BATCH / 16);
    teacher_mlp<<<grid, 128, 0, stream>>>(obs, counts, lists,
                                          w0, b0, w2, b2, w4, b4, w6, b6, out);
}